// HyConv_30648886624885
// MI455X (gfx1250) — compile-verified
//
#include <hip/hip_runtime.h>
#include <hip/hip_bf16.h>

typedef __attribute__((ext_vector_type(2))) float v2f;
typedef __attribute__((ext_vector_type(8))) float v8f;

// ---------------------------------------------------------------------------
// Kernel 1: zero a float buffer (Y accumulator lives in poisoned ws)
// ---------------------------------------------------------------------------
__global__ void hy_zero_f32(float* __restrict__ p, long long n) {
    long long i = (long long)blockIdx.x * blockDim.x + threadIdx.x;
    if (i < n) p[i] = 0.0f;
}

// ---------------------------------------------------------------------------
// Kernel 2: init output with broadcast bias: out[n*OUT + c] = bias[c]
// ---------------------------------------------------------------------------
__global__ void hy_init_bias(float* __restrict__ out, const float* __restrict__ bias,
                             long long total, int out_dim) {
    long long i = (long long)blockIdx.x * blockDim.x + threadIdx.x;
    if (i < total) out[i] = bias[(int)(i % out_dim)];
}

// ---------------------------------------------------------------------------
// Kernel 3: Xt = X @ theta via V_WMMA_F32_16X16X4_F32.
// One wave32 per 16x16 output tile. K is looped in steps of 4.
// A (16x4 f32) layout: lanes 0-15 -> M=lane, VGPR0=K0, VGPR1=K1;
//                      lanes 16-31 -> M=lane-16, VGPR0=K2, VGPR1=K3.
// B (4x16 f32) layout: row striped across lanes within a VGPR:
//                      VGPR0: lanes 0-15 = K0 row, lanes 16-31 = K2 row;
//                      VGPR1: lanes 0-15 = K1 row, lanes 16-31 = K3 row.
// C/D (16x16 f32): VGPR r, lanes 0-15 -> M=r, lanes 16-31 -> M=r+8, N=lane%16.
// ---------------------------------------------------------------------------
__global__ void hy_gemm_wmma(const float* __restrict__ X,     // [N, K] row-major
                             const float* __restrict__ T,     // [K, M] row-major
                             float* __restrict__ Xt,          // [N, M] row-major
                             int N, int K, int M) {
    int wave  = (int)((blockIdx.x * (long long)blockDim.x + threadIdx.x) >> 5);
    int lane  = threadIdx.x & 31;
    int tilesN = M >> 4;                 // 128/16 = 8
    int tileM  = (wave / tilesN) << 4;
    int tileN  = (wave % tilesN) << 4;
    if (tileM >= N) return;

    int lsub = lane & 15;                // position within half-wave
    int hi   = lane >> 4;                // 0: K pair {0,1}; 1: K pair {2,3}

    const float* arow = X + (long long)(tileM + lsub) * K + 2 * hi;
    const float* bcol = T + (long long)(2 * hi) * M + (tileN + lsub);

    v8f c = {};
    for (int k = 0; k < K; k += 4) {
        v2f a = *(const v2f*)(arow + k);            // K = k+2*hi, k+2*hi+1 (8B aligned)
        v2f b;
        b.x = bcol[(long long)k * M];               // theta[k+2*hi    ][tileN+lsub]
        b.y = bcol[(long long)(k + 1) * M];         // theta[k+2*hi + 1][tileN+lsub]
        c = __builtin_amdgcn_wmma_f32_16x16x4_f32(
                /*neg_a=*/false, a, /*neg_b=*/false, b,
                /*c_mod=*/(short)0, c, /*reuse_a=*/false, /*reuse_b=*/false);
    }

    int row0 = tileM + hi * 8;
    int col  = tileN + lsub;
    #pragma unroll
    for (int r = 0; r < 8; ++r)
        Xt[(long long)(row0 + r) * M + col] = c[r];
}

// ---------------------------------------------------------------------------
// Kernels 4/5: scatter-add of 128-float rows. One wave32 per nnz entry,
// each lane moves a float4 (global_load_b128 gather + 4 f32 atomics).
// All hot data (Xt 25.6MB, Y 5.1MB, out 25.6MB) is L2-resident on MI455X.
// ---------------------------------------------------------------------------
__global__ void hy_scatter_add(const float* __restrict__ src,      // [S, 128]
                               const int*   __restrict__ gather_idx,
                               const int*   __restrict__ scatter_idx,
                               float* __restrict__ dst,            // [D, 128]
                               int nnz) {
    long long gid = (long long)blockIdx.x * blockDim.x + threadIdx.x;
    int e    = (int)(gid >> 5);
    int lane = (int)(gid & 31);
    if (e >= nnz) return;

    int s = gather_idx[e];
    int d = scatter_idx[e];

    const float4 v = *(const float4*)(src + (long long)s * 128 + lane * 4);
    float* dp = dst + (long long)d * 128 + lane * 4;
    unsafeAtomicAdd(dp + 0, v.x);
    unsafeAtomicAdd(dp + 1, v.y);
    unsafeAtomicAdd(dp + 2, v.z);
    unsafeAtomicAdd(dp + 3, v.w);
}

// ---------------------------------------------------------------------------
extern "C" void kernel_launch(void* const* d_in, const int* in_sizes, int n_in,
                              void* d_out, int out_size, void* d_ws, size_t ws_size,
                              hipStream_t stream) {
    const float* X          = (const float*)d_in[0];
    const float* theta      = (const float*)d_in[1];
    const float* bias       = (const float*)d_in[2];
    const int*   node_idx   = (const int*)d_in[3];
    const int*   hyedge_idx = (const int*)d_in[4];

    const int OUT   = in_sizes[2];             // 128
    const int IN    = in_sizes[1] / OUT;       // 256
    const int N     = in_sizes[0] / IN;        // 50000
    const int NNZ   = in_sizes[3];             // 1,600,000
    const int EDGES = 10000;                   // N_EDGES (reference constant)

    float* Xt = (float*)d_ws;                          // [N, OUT]  25.6 MB
    float* Y  = Xt + (size_t)N * OUT;                  // [EDGES, OUT] 5.12 MB

    // 1) zero hyperedge accumulator
    {
        long long n = (long long)EDGES * OUT;
        int blocks = (int)((n + 255) / 256);
        hy_zero_f32<<<blocks, 256, 0, stream>>>(Y, n);
    }
    // 2) Xt = X @ theta  (WMMA f32)
    {
        long long waves   = (long long)(N / 16) * (OUT / 16);   // 25000
        long long threads = waves * 32;
        int blocks = (int)((threads + 255) / 256);
        hy_gemm_wmma<<<blocks, 256, 0, stream>>>(X, theta, Xt, N, IN, OUT);
    }
    // 3) out = broadcast(bias)
    {
        long long n = (long long)N * OUT;
        int blocks = (int)((n + 255) / 256);
        hy_init_bias<<<blocks, 256, 0, stream>>>((float*)d_out, bias, n, OUT);
    }
    // 4) Y[hyedge_idx[e]] += Xt[node_idx[e]]
    {
        long long threads = (long long)NNZ * 32;
        int blocks = (int)((threads + 255) / 256);
        hy_scatter_add<<<blocks, 256, 0, stream>>>(Xt, node_idx, hyedge_idx, Y, NNZ);
    }
    // 5) out[node_idx[e]] += Y[hyedge_idx[e]]
    {
        long long threads = (long long)NNZ * 32;
        int blocks = (int)((threads + 255) / 256);
        hy_scatter_add<<<blocks, 256, 0, stream>>>(Y, hyedge_idx, node_idx, (float*)d_out, NNZ);
    }
}